// GCNBaseline_80513456930966
// MI455X (gfx1250) — compile-verified
//
#include <hip/hip_runtime.h>

// CDNA5 fp32 WMMA fragment types: 16x4 A / 4x16 B = 2 VGPRs/lane, 16x16 C/D = 8 VGPRs/lane (wave32)
typedef __attribute__((ext_vector_type(2))) float v2f;
typedef __attribute__((ext_vector_type(8))) float v8f;

// ---------------- utility kernels ----------------
__global__ __launch_bounds__(256) void gcn_fill_f32(float* __restrict__ p, float v, int n) {
    int i = blockIdx.x * 256 + threadIdx.x;
    if (i < n) p[i] = v;
}

__global__ __launch_bounds__(256) void gcn_zero_f4(float4* __restrict__ p, int n4) {
    int i = blockIdx.x * 256 + threadIdx.x;
    if (i < n4) p[i] = make_float4(0.f, 0.f, 0.f, 0.f);
}

__global__ __launch_bounds__(256) void gcn_deg_accum(const int* __restrict__ dst,
                                                     const float* __restrict__ ew,
                                                     float* __restrict__ deg, int nE) {
    int e = blockIdx.x * 256 + threadIdx.x;
    if (e < nE) atomicAdd(&deg[dst[e]], ew[e]);
}

__global__ __launch_bounds__(256) void gcn_deg_to_dinv(float* __restrict__ deg, int n) {
    int i = blockIdx.x * 256 + threadIdx.x;
    if (i < n) {
        float d = deg[i];
        deg[i] = (d > 0.f) ? rsqrtf(d) : 0.f;
    }
}

// ---------------- WMMA fp32 GEMM: Y[nRows x 64] = X[nRows x 64] @ W[64 x 64] ----------------
#define XPAD 66      // 64 + 2: stride-64 column reads would be 16-way bank conflicts otherwise
#define WSTRIDE 80   // pair-row stride; 2*WSTRIDE % 64 == 32 -> half-waves hit disjoint bank groups

__global__ __launch_bounds__(256)
void gcn_gemm64_wmma(const float* __restrict__ X, const float* __restrict__ W,
                     float* __restrict__ Y, int nRows) {
    __shared__ float sX[128 * XPAD];          // 128-row x 64-col input tile
    __shared__ float sW[32 * WSTRIDE * 2];    // W as (kpair, col, 2) interleaved pairs

    const int tid  = threadIdx.x;
    const int lane = tid & 31;
    const int wave = tid >> 5;                // 8 waves, 16 rows each
    const int row0 = blockIdx.x * 128;

    // Stage W: pair rows (2p, 2p+1) interleaved per column -> one ds_load_b64 per B fragment
    for (int i = tid; i < 32 * 64; i += 256) {
        int p = i >> 6;
        int c = i & 63;
        sW[(p * WSTRIDE + c) * 2 + 0] = W[(2 * p + 0) * 64 + c];
        sW[(p * WSTRIDE + c) * 2 + 1] = W[(2 * p + 1) * 64 + c];
    }
    // Stage X tile with b128 loads
    for (int i = tid; i < 128 * 16; i += 256) {
        int r  = i >> 4;
        int c4 = (i & 15) << 2;
        int gr = row0 + r;
        float4 v = make_float4(0.f, 0.f, 0.f, 0.f);
        if (gr < nRows) v = *(const float4*)(X + (size_t)gr * 64 + c4);
        float* dp = &sX[r * XPAD + c4];
        dp[0] = v.x; dp[1] = v.y; dp[2] = v.z; dp[3] = v.w;
    }
    __syncthreads();

    // ISA 16x4 f32 A layout: lanes 0-15 hold K=k..k+1, lanes 16-31 hold K=k+2..k+3 (row = lane&15)
    const int mrow  = lane & 15;
    const int khalf = (lane >> 4) << 1;       // 0 or 2
    const int col   = lane & 15;              // B/D column within 16-col group
    v8f acc0 = {}, acc1 = {}, acc2 = {}, acc3 = {};
    const float* xrow = &sX[(wave * 16 + mrow) * XPAD];

#pragma unroll
    for (int k = 0; k < 64; k += 4) {
        const int kk = k + khalf;             // even -> float2 aligned
        v2f a = *(const v2f*)(xrow + kk);
        const float* wp = &sW[((kk >> 1) * WSTRIDE) * 2];
        v2f b0 = *(const v2f*)(wp + (col +  0) * 2);
        v2f b1 = *(const v2f*)(wp + (col + 16) * 2);
        v2f b2 = *(const v2f*)(wp + (col + 32) * 2);
        v2f b3 = *(const v2f*)(wp + (col + 48) * 2);
        acc0 = __builtin_amdgcn_wmma_f32_16x16x4_f32(false, a, false, b0, (short)0, acc0, false, false);
        acc1 = __builtin_amdgcn_wmma_f32_16x16x4_f32(false, a, false, b1, (short)0, acc1, false, false);
        acc2 = __builtin_amdgcn_wmma_f32_16x16x4_f32(false, a, false, b2, (short)0, acc2, false, false);
        acc3 = __builtin_amdgcn_wmma_f32_16x16x4_f32(false, a, false, b3, (short)0, acc3, false, false);
    }

    // D layout: VGPR i -> row i (lanes 0-15) / row i+8 (lanes 16-31), col = lane&15
    const int rOff = (lane >> 4) << 3;
#pragma unroll
    for (int i = 0; i < 8; ++i) {
        int gr = row0 + wave * 16 + rOff + i;
        if (gr < nRows) {
            float* yp = Y + (size_t)gr * 64 + col;
            yp[0] = acc0[i]; yp[16] = acc1[i]; yp[32] = acc2[i]; yp[48] = acc3[i];
        }
    }
}

// ---------------- edge scatter: AGG[d] += dinv[s]*w*dinv[d] * HL[s] ----------------
__global__ __launch_bounds__(256)
void gcn_scatter(const float* __restrict__ HL, float* __restrict__ AGG,
                 const int* __restrict__ src, const int* __restrict__ dst,
                 const float* __restrict__ ew, const float* __restrict__ dinv, int nE) {
    const int lane = threadIdx.x & 31;
    const int wave = threadIdx.x >> 5;
    int e = blockIdx.x * 8 + wave;            // one wave32 per edge
    if (e >= nE) return;
    int s = src[e];
    int d = dst[e];
    float c = dinv[s] * ew[e] * dinv[d];
    float2 v = *(const float2*)(HL + (size_t)s * 64 + lane * 2);  // 2 features/lane
    float* op = AGG + (size_t)d * 64 + lane * 2;
    atomicAdd(op + 0, c * v.x);
    atomicAdd(op + 1, c * v.y);
}

// ---------------- self-loop + bias + relu (in place on AGG) ----------------
__global__ __launch_bounds__(256)
void gcn_finish(const float* __restrict__ HL, float* __restrict__ AGG,
                const float* __restrict__ dinv, const float* __restrict__ b, int nNodes) {
    int i = blockIdx.x * 256 + threadIdx.x;   // one float4 per thread
    int n = i >> 4;
    if (n >= nNodes) return;
    int f4 = (i & 15) << 2;
    float di = dinv[n];
    float w  = di * di;                        // self-loop coefficient dinv^2 * 1.0
    float4 hl = *(const float4*)(HL + (size_t)n * 64 + f4);
    float4 ag = *(const float4*)(AGG + (size_t)n * 64 + f4);
    float4 bb = *(const float4*)(b + f4);
    float4 r;
    r.x = fmaxf(ag.x + w * hl.x + bb.x, 0.f);
    r.y = fmaxf(ag.y + w * hl.y + bb.y, 0.f);
    r.z = fmaxf(ag.z + w * hl.z + bb.z, 0.f);
    r.w = fmaxf(ag.w + w * hl.w + bb.w, 0.f);
    *(float4*)(AGG + (size_t)n * 64 + f4) = r;
}

// ---------------- head: OUT[nRows x 12] = H[nRows x 64] @ Wfc[64 x 12] + bfc ----------------
__global__ __launch_bounds__(256)
void gcn_gemm_out_wmma(const float* __restrict__ H, const float* __restrict__ Wfc,
                       const float* __restrict__ bfc, float* __restrict__ Y, int nRows) {
    const int lane = threadIdx.x & 31;
    const int wave = threadIdx.x >> 5;
    const int row0 = blockIdx.x * 128 + wave * 16;
    const int mrow  = lane & 15;
    const int khalf = (lane >> 4) << 1;
    const int col   = lane & 15;
    v8f acc = {};
#pragma unroll
    for (int k = 0; k < 64; k += 4) {
        const int kk = k + khalf;
        v2f a = {0.f, 0.f};
        int gr = row0 + mrow;
        if (gr < nRows) a = *(const v2f*)(H + (size_t)gr * 64 + kk);
        v2f b = {0.f, 0.f};
        if (col < 12) {                        // pad 12 cols -> 16 with zeros
            b.x = Wfc[kk * 12 + col];
            b.y = Wfc[(kk + 1) * 12 + col];
        }
        acc = __builtin_amdgcn_wmma_f32_16x16x4_f32(false, a, false, b, (short)0, acc, false, false);
    }
    if (col < 12) {
        float bias = bfc[col];
        const int rOff = (lane >> 4) << 3;
#pragma unroll
        for (int i = 0; i < 8; ++i) {
            int gr = row0 + rOff + i;
            if (gr < nRows) Y[(size_t)gr * 12 + col] = acc[i] + bias;
        }
    }
}

// ---------------- launch ----------------
extern "C" void kernel_launch(void* const* d_in, const int* in_sizes, int n_in,
                              void* d_out, int out_size, void* d_ws, size_t ws_size,
                              hipStream_t stream) {
    const float* x   = (const float*)d_in[0];
    const int*   ei  = (const int*)d_in[1];
    const float* ea  = (const float*)d_in[2];
    const float* W1  = (const float*)d_in[3];
    const float* b1  = (const float*)d_in[4];
    const float* W2  = (const float*)d_in[5];
    const float* b2  = (const float*)d_in[6];
    const float* Wfc = (const float*)d_in[7];
    const float* bfc = (const float*)d_in[8];
    float* out = (float*)d_out;

    const int N = in_sizes[0] / 64;           // 50000
    const int E = in_sizes[2];                // 800000 (edge_attr count)
    const int* src = ei;                      // edge_index[0]
    const int* dst = ei + E;                  // edge_index[1]

    char* ws = (char*)d_ws;
    float* dinv = (float*)ws;                 // N floats (deg -> dinv in place)
    float* buf0 = (float*)(ws + (size_t)(1u  << 20));   // N x 64: linear output hl
    float* buf1 = (float*)(ws + (size_t)(17u << 20));   // N x 64: aggregate / h

    const int nf4 = (N * 64) / 4;
    dim3 blk(256);
    int gN   = (N + 255) / 256;
    int gE   = (E + 255) / 256;
    int gF4  = (nf4 + 255) / 256;
    int gNF  = (N * 16 + 255) / 256;
    int gRow = (N + 127) / 128;
    int gEW  = (E + 7) / 8;                   // 8 edge-waves per block

    // degrees (self-loop weight 1.0 folded into init) and dinv = rsqrt(deg)
    gcn_fill_f32<<<gN, blk, 0, stream>>>(dinv, 1.0f, N);
    gcn_deg_accum<<<gE, blk, 0, stream>>>(dst, ea, dinv, E);
    gcn_deg_to_dinv<<<gN, blk, 0, stream>>>(dinv, N);

    // layer 1: hl = x@W1 ; agg = scatter ; h1 = relu(agg + dinv^2*hl + b1)
    gcn_gemm64_wmma<<<gRow, blk, 0, stream>>>(x, W1, buf0, N);
    gcn_zero_f4<<<gF4, blk, 0, stream>>>((float4*)buf1, nf4);
    gcn_scatter<<<gEW, blk, 0, stream>>>(buf0, buf1, src, dst, ea, dinv, E);
    gcn_finish<<<gNF, blk, 0, stream>>>(buf0, buf1, dinv, b1, N);

    // layer 2 (buf1 free for reuse once the GEMM has consumed it — stream ordered)
    gcn_gemm64_wmma<<<gRow, blk, 0, stream>>>(buf1, W2, buf0, N);
    gcn_zero_f4<<<gF4, blk, 0, stream>>>((float4*)buf1, nf4);
    gcn_scatter<<<gEW, blk, 0, stream>>>(buf0, buf1, src, dst, ea, dinv, E);
    gcn_finish<<<gNF, blk, 0, stream>>>(buf0, buf1, dinv, b2, N);

    // head
    gcn_gemm_out_wmma<<<gRow, blk, 0, stream>>>(buf1, Wfc, bfc, out, N);
}